// ClassicalMappedQRNN_41386304864597
// MI455X (gfx1250) — compile-verified
//
#include <hip/hip_runtime.h>
#include <math.h>

// ---------------------------------------------------------------------------
// ClassicalMappedQRNN scan: B=8192 independent rows, S=4096 sequential steps.
// TWO rows per lane (interleaved independent recurrences) to hide the
// dot->v_rsq->fma dependency chain — the workload is latency-bound (only
// 128 waves machine-wide), not bandwidth-bound (134 MB @ 23.3 TB/s ~ 6us).
// Wave-private double-buffered async global->LDS b128 tiles (gfx1250
// ASYNCcnt path) give coalesced traffic despite the 16KB row stride.
// Deferred normalization: carry unnormalized g + ip = 1/||g||.
// ---------------------------------------------------------------------------

#define TS 32                          // timesteps per tile
#define ROWSTRIDE 36                   // 36 floats = 144B: 16B-aligned rows; b128 LDS reads
                                       // hit each of the 64 banks exactly twice (optimal);
                                       // row l+32 offset = 1152 dwords = 0 mod 64 -> same pattern
#define ROWS_PER_WAVE 64               // 2 rows per lane
#define WAVES_PER_BLOCK 2
#define BLOCK_THREADS (WAVES_PER_BLOCK * 32)
#define ROWS_PER_BLOCK (WAVES_PER_BLOCK * ROWS_PER_WAVE)
#define TILE_FLOATS (ROWS_PER_WAVE * ROWSTRIDE)   // 2304 floats per wave-buffer

#if defined(__HIP_DEVICE_COMPILE__) && __has_builtin(__builtin_amdgcn_global_load_async_to_lds_b64)
  #define HAVE_ASYNC_LDS 1
#else
  #define HAVE_ASYNC_LDS 0
#endif

#if defined(__HIP_DEVICE_COMPILE__) && __has_builtin(__builtin_amdgcn_s_wait_asynccnt)
  #define WAIT_ASYNCCNT(n) __builtin_amdgcn_s_wait_asynccnt(n)
#elif defined(__HIP_DEVICE_COMPILE__) && HAVE_ASYNC_LDS
  #define WAIT_ASYNCCNT(n) asm volatile("s_wait_asynccnt %0" :: "i"(n) : "memory")
#else
  #define WAIT_ASYNCCNT(n) do {} while (0)
#endif

#if HAVE_ASYNC_LDS
typedef int v2i __attribute__((vector_size(2 * sizeof(int))));
typedef int v4i __attribute__((vector_size(4 * sizeof(int))));
typedef __attribute__((address_space(1))) v2i* global_v2i_ptr;
typedef __attribute__((address_space(3))) v2i* local_v2i_ptr;
typedef __attribute__((address_space(1))) v4i* global_v4i_ptr;
typedef __attribute__((address_space(3))) v4i* local_v4i_ptr;

__device__ __forceinline__ void async_copy_b64(const float* gp, float* lp) {
    __builtin_amdgcn_global_load_async_to_lds_b64(
        (global_v2i_ptr)gp, (local_v2i_ptr)lp, /*imm offset*/0, /*cpol*/0);
}

#if __has_builtin(__builtin_amdgcn_global_load_async_to_lds_b128)
  #define ASYNCS_PER_TILE 16           // 16 x 512B per wave-tile (max 32 outstanding < 63)
  __device__ __forceinline__ void async_copy_b128(const float* gp, float* lp) {
      __builtin_amdgcn_global_load_async_to_lds_b128(
          (global_v4i_ptr)gp, (local_v4i_ptr)lp, /*imm offset*/0, /*cpol*/0);
  }
#else
  #define ASYNCS_PER_TILE 32           // b64 fallback: 32 per tile (HW stalls at cnt cap, ok)
  __device__ __forceinline__ void async_copy_b128(const float* gp, float* lp) {
      async_copy_b64(gp, lp);
      async_copy_b64(gp + 2, lp + 2);
  }
#endif
#endif // HAVE_ASYNC_LDS

// Hardware trig: v_cos_f32 / v_sin_f32 take input in revolutions (x * 2*pi rad).
__device__ __forceinline__ float hw_cos(float theta) {
#if defined(__HIP_DEVICE_COMPILE__)
    return __builtin_amdgcn_cosf(theta * 0.15915494309189535f);
#else
    return cosf(theta);
#endif
}
__device__ __forceinline__ float hw_sin(float theta) {
#if defined(__HIP_DEVICE_COMPILE__)
    return __builtin_amdgcn_sinf(theta * 0.15915494309189535f);
#else
    return sinf(theta);
#endif
}

// One scan step with deferred normalization.
// State: g (unnormalized hidden, == h * ||g||) and ip = 1/||g|| (ip=0 at t=0).
__device__ __forceinline__ void qstep(float x,
                                      float& g0, float& g1, float& g2, float& g3,
                                      float& ip,
                                      float ca, float sa, float cb, float sb) {
    // phi = atan(x); c = cos(phi/2), s = sin(phi/2) via half-angle:
    //   r = cos(phi) = 1/sqrt(1+x^2);  c = sqrt((1+r)/2);  s = sign(x)*sqrt((1-r)/2)
#if defined(__HIP_DEVICE_COMPILE__)
    float r = __builtin_amdgcn_rsqf(fmaf(x, x, 1.0f));
    float c = __builtin_amdgcn_sqrtf(0.5f * (1.0f + r));
    float s = copysignf(__builtin_amdgcn_sqrtf(fmaxf(0.0f, 0.5f * (1.0f - r))), x);
#else
    float r = 1.0f / sqrtf(fmaf(x, x, 1.0f));
    float c = sqrtf(0.5f * (1.0f + r));
    float s = copysignf(sqrtf(fmaxf(0.0f, 0.5f * (1.0f - r))), x);
#endif
    // u = embedded @ rx.T = (ca*c, -sa*s, ca*s, sa*c)   [unit norm, off-chain]
    float u0 =  ca * c;
    float u1 = -sa * s;
    float u2 =  ca * s;
    float u3 =  sa * c;
    // rot = rz @ g : two independent 2D rotations (computable before ip arrives)
    float rot0 = fmaf( cb, g0, (-sb) * g1);
    float rot1 = fmaf( sb, g0,   cb  * g1);
    float rot2 = fmaf( cb, g2,   sb  * g3);
    float rot3 = fmaf(-sb, g2,   cb  * g3);
    // fold previous step's normalization in as one FMA per component
    g0 = fmaf(ip, rot0, u0);
    g1 = fmaf(ip, rot1, u1);
    g2 = fmaf(ip, rot2, u2);
    g3 = fmaf(ip, rot3, u3);
    float n2 = fmaf(g0, g0, g1 * g1) + fmaf(g2, g2, g3 * g3);
#if defined(__HIP_DEVICE_COMPILE__)
    ip = __builtin_amdgcn_rsqf(n2);
#else
    ip = 1.0f / sqrtf(n2);
#endif
}

__global__ __launch_bounds__(BLOCK_THREADS)
void qrnn_scan_kernel(const float* __restrict__ x,
                      const float* __restrict__ alpha_p,
                      const float* __restrict__ beta_p,
                      float* __restrict__ out,
                      int B, int S) {
    const int lane   = threadIdx.x & 31;
    const int wave   = threadIdx.x >> 5;
    const int waveB0 = blockIdx.x * ROWS_PER_BLOCK + wave * ROWS_PER_WAVE;
    const int bA = waveB0 + lane;        // lane's first row
    const int bB = waveB0 + 32 + lane;   // lane's second row (independent recurrence)
    if (bB >= B) return;                 // grid sized exactly; kept for safety

    const float alpha = alpha_p[0];
    const float beta  = beta_p[0];
    const float ca = hw_cos(0.5f * alpha), sa = hw_sin(0.5f * alpha);
    const float cb = hw_cos(0.5f * beta),  sb = hw_sin(0.5f * beta);

    // Two interleaved states: while A waits on v_rsq, B's FMAs issue.
    float a0 = 0.0f, a1 = 0.0f, a2 = 0.0f, a3 = 0.0f, ipa = 0.0f;
    float b0 = 0.0f, b1 = 0.0f, b2 = 0.0f, b3 = 0.0f, ipb = 0.0f;

#if HAVE_ASYNC_LDS
    // 2 waves * 2 buffers * 2304 floats = 36864 B LDS, wave-private tiles.
    __shared__ __align__(16) float lds[WAVES_PER_BLOCK * 2 * TILE_FLOATS];
    float* const wbase = &lds[wave * 2 * TILE_FLOATS];

    const int subrow = lane >> 3;        // 0..3: 4 rows per b128 async instruction
    const int col    = (lane & 7) << 2;  // float offset; 8 lanes x 16B = one 128B row
    const int ntiles = S / TS;           // S=4096 -> 128 tiles

    auto issue_tile = [&](int t, int buf) {
        const float* gx = x + (size_t)waveB0 * (size_t)S + (size_t)t * TS;
        float* lbuf = wbase + buf * TILE_FLOATS;
        #pragma unroll
        for (int rr = 0; rr < 16; ++rr) {            // 16 x (4 rows x 128B) = 64-row tile
            const int row = (rr << 2) + subrow;
            const float* gp = gx + (size_t)row * (size_t)S + col;   // contiguous per 8 lanes
            float* lp = lbuf + row * ROWSTRIDE + col;               // 16B-aligned LDS dest
            async_copy_b128(gp, lp);
        }
    };

    auto process_tile = [&](int buf) {
        const float4* rowA = (const float4*)(wbase + buf * TILE_FLOATS + lane * ROWSTRIDE);
        const float4* rowB = (const float4*)(wbase + buf * TILE_FLOATS + (lane + 32) * ROWSTRIDE);
        #pragma unroll
        for (int k = 0; k < TS / 4; ++k) {           // 2x8 ds_load_b128, 2 bank-cycles each
            float4 va = rowA[k];
            float4 vb = rowB[k];
            qstep(va.x, a0, a1, a2, a3, ipa, ca, sa, cb, sb);
            qstep(vb.x, b0, b1, b2, b3, ipb, ca, sa, cb, sb);
            qstep(va.y, a0, a1, a2, a3, ipa, ca, sa, cb, sb);
            qstep(vb.y, b0, b1, b2, b3, ipb, ca, sa, cb, sb);
            qstep(va.z, a0, a1, a2, a3, ipa, ca, sa, cb, sb);
            qstep(vb.z, b0, b1, b2, b3, ipb, ca, sa, cb, sb);
            qstep(va.w, a0, a1, a2, a3, ipa, ca, sa, cb, sb);
            qstep(vb.w, b0, b1, b2, b3, ipb, ca, sa, cb, sb);
        }
    };

    issue_tile(0, 0);
    for (int t = 0; t < ntiles - 1; ++t) {          // steady state: prefetch pipeline
        issue_tile(t + 1, (t + 1) & 1);             // max 2*ASYNCS_PER_TILE outstanding
        WAIT_ASYNCCNT(ASYNCS_PER_TILE);             // asyncs retire in order -> tile t resident
        asm volatile("" ::: "memory");
        process_tile(t & 1);
    }
    WAIT_ASYNCCNT(0);                               // drain: last tile resident
    asm volatile("" ::: "memory");
    process_tile((ntiles - 1) & 1);
#else
    // Fallback: per-lane float4 streaming for both rows
    const float4* xrowA = (const float4*)(x + (size_t)bA * (size_t)S);
    const float4* xrowB = (const float4*)(x + (size_t)bB * (size_t)S);
    for (int k = 0; k < (S >> 2); ++k) {
        float4 va = xrowA[k];
        float4 vb = xrowB[k];
        qstep(va.x, a0, a1, a2, a3, ipa, ca, sa, cb, sb);
        qstep(vb.x, b0, b1, b2, b3, ipb, ca, sa, cb, sb);
        qstep(va.y, a0, a1, a2, a3, ipa, ca, sa, cb, sb);
        qstep(vb.y, b0, b1, b2, b3, ipb, ca, sa, cb, sb);
        qstep(va.z, a0, a1, a2, a3, ipa, ca, sa, cb, sb);
        qstep(vb.z, b0, b1, b2, b3, ipb, ca, sa, cb, sb);
        qstep(va.w, a0, a1, a2, a3, ipa, ca, sa, cb, sb);
        qstep(vb.w, b0, b1, b2, b3, ipb, ca, sa, cb, sb);
    }
#endif

    // h = g*ip;  z = (h0^2+h1^2) - (h2^2+h3^2) = (pa - pb) * ip^2 per row
    float paA = fmaf(a0, a0, a1 * a1);
    float pbA = fmaf(a2, a2, a3 * a3);
    out[bA] = (paA - pbA) * (ipa * ipa);
    float paB = fmaf(b0, b0, b1 * b1);
    float pbB = fmaf(b2, b2, b3 * b3);
    out[bB] = (paB - pbB) * (ipb * ipb);
}

extern "C" void kernel_launch(void* const* d_in, const int* in_sizes, int n_in,
                              void* d_out, int out_size, void* d_ws, size_t ws_size,
                              hipStream_t stream) {
    (void)n_in; (void)d_ws; (void)ws_size;
    const float* x       = (const float*)d_in[0];
    const float* alpha_p = (const float*)d_in[1];
    const float* beta_p  = (const float*)d_in[2];
    float* out = (float*)d_out;

    const int B = out_size;              // 8192
    const int S = in_sizes[0] / B;       // 4096

    const int grid = (B + ROWS_PER_BLOCK - 1) / ROWS_PER_BLOCK;   // 64 blocks of 2 waves
    qrnn_scan_kernel<<<grid, BLOCK_THREADS, 0, stream>>>(x, alpha_p, beta_p, out, B, S);
}